// YoloV1Loss_46024869544379
// MI455X (gfx1250) — compile-verified
//
#include <hip/hip_runtime.h>

#define SS 7
#define NBOX 2
#define NCLS 20
#define NGT 16
#define CELLS (SS * SS)
#define CH (NBOX * 5 + NCLS)     // 30
#define SLICE (CELLS * CH)       // 1470 floats per batch element
#define NTHREADS 128

// ---- CDNA5 async global->LDS path (guarded; falls back to sync copy) ----
#if defined(__HIP_DEVICE_COMPILE__) && defined(__gfx1250__) && \
    __has_builtin(__builtin_amdgcn_global_load_async_to_lds_b64)
#define HAVE_ASYNC_LDS 1
#else
#define HAVE_ASYNC_LDS 0
#endif

#if HAVE_ASYNC_LDS && __has_builtin(__builtin_amdgcn_global_load_async_to_lds_b128)
#define HAVE_ASYNC_LDS_B128 1
#else
#define HAVE_ASYNC_LDS_B128 0
#endif

typedef int v2i __attribute__((vector_size(2 * sizeof(int))));
typedef int v4i __attribute__((vector_size(4 * sizeof(int))));
typedef __attribute__((address_space(1))) v2i GV2;  // global (AS1)
typedef __attribute__((address_space(3))) v2i LV2;  // LDS (AS3)
typedef __attribute__((address_space(1))) v4i GV4;
typedef __attribute__((address_space(3))) v4i LV4;

__device__ __forceinline__ void async_copy_b64(const float* g, float* l) {
#if HAVE_ASYNC_LDS
  __builtin_amdgcn_global_load_async_to_lds_b64(
      (GV2*)(uintptr_t)g, (LV2*)l, /*offset=*/0, /*cpol=*/0);
#else
  l[0] = g[0];
  l[1] = g[1];
#endif
}

__device__ __forceinline__ void async_copy_b128(const float* g, float* l) {
#if HAVE_ASYNC_LDS_B128
  __builtin_amdgcn_global_load_async_to_lds_b128(
      (GV4*)(uintptr_t)g, (LV4*)l, /*offset=*/0, /*cpol=*/0);
#else
  async_copy_b64(g, l);
  async_copy_b64(g + 2, l + 2);
#endif
}

__device__ __forceinline__ void wait_async_lds() {
#if HAVE_ASYNC_LDS
#if __has_builtin(__builtin_amdgcn_s_wait_asynccnt)
  __builtin_amdgcn_s_wait_asynccnt(0);
#else
  asm volatile("s_wait_asynccnt 0" ::: "memory");
#endif
#endif
}

// One block per batch element.
__global__ __launch_bounds__(NTHREADS) void yolo_loss_kernel(
    const float* __restrict__ outputs,     // (batch, 7, 7, 30)
    const float* __restrict__ gt_boxes,    // (batch, 16, 4)
    const long long* __restrict__ labels,  // (batch, 16) int64 in [1, 20]
    float* __restrict__ partials) {
  __shared__ __align__(16) float s_out[SLICE];
  __shared__ __align__(16) float s_gt[NGT * 4];
  __shared__ float s_cls[CELLS];
  __shared__ float s_red[NTHREADS];
  __shared__ unsigned s_mask;

  const int tid = threadIdx.x;
  const int b = blockIdx.x;

  if (tid == 0) s_mask = 0u;

  // Stage the 5880-byte output slice: 367 x b128 chunks + one b64 tail.
  const float* gsrc = outputs + (size_t)b * SLICE;
#pragma unroll
  for (int k = 0; k < 3; ++k) {
    int i = tid + k * NTHREADS;                 // b128 chunk index
    if (i < SLICE / 4) async_copy_b128(gsrc + 4 * i, s_out + 4 * i);
  }
  if (tid == NTHREADS - 1)                      // tail floats [1468, 1470)
    async_copy_b64(gsrc + (SLICE & ~3), s_out + (SLICE & ~3));
  // Stage 16 GT boxes (64 floats = 16 b128 chunks).
  if (tid < NGT)
    async_copy_b128(gt_boxes + (size_t)b * NGT * 4 + 4 * tid, s_gt + 4 * tid);

  long long mylab = 1;
  if (tid < NGT) mylab = labels[(size_t)b * NGT + tid];

  wait_async_lds();
  __syncthreads();  // s_out / s_gt complete; s_mask init visible

  if (tid < NGT) atomicOr(&s_mask, 1u << ((int)mylab - 1));
  __syncthreads();

  const unsigned mask = s_mask;
  const float K = (float)__popc(mask);

  // Phase A: per-cell class loss  cls = K*logsumexp(pc) - sum(masked pc)
  if (tid < CELLS) {
    const float* pc = s_out + tid * CH + NBOX * 5;
    float m = pc[0];
#pragma unroll
    for (int c = 1; c < NCLS; ++c) m = fmaxf(m, pc[c]);
    float sum = 0.f, dot = 0.f;
#pragma unroll
    for (int c = 0; c < NCLS; ++c) {
      float v = pc[c];
      sum += expf(v - m);
      if (mask & (1u << c)) dot += v;
    }
    s_cls[tid] = K * (m + logf(sum)) - dot;
  }
  __syncthreads();

  // Phase B: per (cell, box) term
  float term = 0.f;
  if (tid < CELLS * NBOX) {
    const int cell = tid >> 1;
    const int box = tid & 1;
    const float* p = s_out + cell * CH + box * 5;
    const float px = p[0], py = p[1], pw = p[2], ph = p[3], conf = p[4];
    const float px2 = px + pw, py2 = py + ph;
    const float parea = pw * ph;

    float best = -1.f;
    int bi = 0;
#pragma unroll
    for (int g = 0; g < NGT; ++g) {
      const float gx = s_gt[4 * g + 0], gy = s_gt[4 * g + 1];
      const float gw = s_gt[4 * g + 2], gh = s_gt[4 * g + 3];
      float iw = fminf(px2, gx + gw) - fmaxf(px, gx);
      float ih = fminf(py2, gy + gh) - fmaxf(py, gy);
      iw = fmaxf(iw, 0.f);
      ih = fmaxf(ih, 0.f);
      const float inter = iw * ih;
      const float uni = parea + gw * gh - inter;
      const float iou = inter / (uni + 1e-10f);
      if (iou > best) { best = iou; bi = g; }  // strict > == argmax first-max
    }

    if (best > 0.f) {
      const float mx = s_gt[4 * bi + 0], my = s_gt[4 * bi + 1];
      const float mw = s_gt[4 * bi + 2], mh = s_gt[4 * bi + 3];
      const int row = cell / SS;
      const int col = cell - row * SS;
      const float x = px + (float)col / (float)SS;
      const float y = py + (float)row / (float)SS;
      const float dx = x - mx, dy = y - my;
      const float dw = sqrtf(pw) - sqrtf(mw);
      const float dh = sqrtf(ph) - sqrtf(mh);
      const float loc = 5.0f * (dx * dx + dy * dy + dw * dw + dh * dh);
      const float dc = conf - best;
      term = loc + dc * dc + s_cls[cell];
    } else {
      term = 0.5f * conf * conf;
    }
  }

  // Deterministic block tree reduction.
  s_red[tid] = term;
  __syncthreads();
#pragma unroll
  for (int off = NTHREADS / 2; off > 0; off >>= 1) {
    if (tid < off) s_red[tid] += s_red[tid + off];
    __syncthreads();
  }
  if (tid == 0) partials[b] = s_red[0];
}

// Deterministic final reduction of per-block partials.
__global__ __launch_bounds__(256) void yolo_reduce_kernel(
    const float* __restrict__ partials, float* __restrict__ out, int n) {
  __shared__ float s[256];
  const int tid = threadIdx.x;
  float acc = 0.f;
  for (int i = tid; i < n; i += 256) acc += partials[i];  // fixed order
  s[tid] = acc;
  __syncthreads();
#pragma unroll
  for (int off = 128; off > 0; off >>= 1) {
    if (tid < off) s[tid] += s[tid + off];
    __syncthreads();
  }
  if (tid == 0) out[0] = s[0] / (float)n;
}

extern "C" void kernel_launch(void* const* d_in, const int* in_sizes, int n_in,
                              void* d_out, int out_size, void* d_ws,
                              size_t ws_size, hipStream_t stream) {
  const float* outputs = (const float*)d_in[0];
  const float* gt_boxes = (const float*)d_in[1];
  const long long* labels = (const long long*)d_in[2];  // int64 per reference
  const int batch = in_sizes[0] / SLICE;                // 16384

  float* partials = (float*)d_ws;  // batch floats of scratch

  yolo_loss_kernel<<<batch, NTHREADS, 0, stream>>>(outputs, gt_boxes, labels,
                                                   partials);
  yolo_reduce_kernel<<<1, 256, 0, stream>>>(partials, (float*)d_out, batch);
}